// GCNII_33217277067913
// MI455X (gfx1250) — compile-verified
//
#include <hip/hip_runtime.h>
#include <math.h>

// ---------------- problem constants (from reference) ----------------
#define NNODES  100000
#define NEDGES  1600000
#define INC     128
#define HID     64
#define OUTC    40
#define NLAYERS 8
#define ALPHA_F 0.1f
#define OMA_F   0.9f            // 1 - ALPHA
#define BN_EPS  1e-5f

typedef float v2f __attribute__((ext_vector_type(2)));
typedef float v8f __attribute__((ext_vector_type(8)));

__device__ __forceinline__ void atomAddF(float* p, float v) {
#if defined(__HIP_DEVICE_COMPILE__)
  unsafeAtomicAdd(p, v);        // hardware global_atomic_add_f32
#else
  (void)p; (void)v;
#endif
}

// ---------------- elementwise helpers ----------------
__global__ void k_fill(float* __restrict__ p, long n, float val) {
  long i = (long)blockIdx.x * blockDim.x + threadIdx.x;
  long st = (long)gridDim.x * blockDim.x;
  for (; i < n; i += st) p[i] = val;
}

__global__ void k_degcount(const int* __restrict__ col, float* __restrict__ deg) {
  int i = blockIdx.x * blockDim.x + threadIdx.x;
  if (i < NEDGES) atomAddF(&deg[col[i]], 1.0f);
}

__global__ void k_rsqrt(float* __restrict__ p) {
  int i = blockIdx.x * blockDim.x + threadIdx.x;
  if (i < NNODES) p[i] = rsqrtf(p[i]);
}

// Wp = (1-beta)*I + beta*convW[l] ; also zero BN stats accumulators
__global__ void k_wp(const float* __restrict__ convWl, float* __restrict__ Wp,
                     float beta, float* __restrict__ stats) {
  int i = blockIdx.x * blockDim.x + threadIdx.x;
  if (i < HID * HID) {
    int k = i >> 6, n = i & 63;
    Wp[i] = beta * convWl[i] + ((k == n) ? (1.0f - beta) : 0.0f);
  }
  if (i < 2 * HID) stats[i] = 0.0f;
}

// agg[col] += (dinv[row]*dinv[col]) * h[row]   (one lane = one edge x 4 channels)
__global__ void k_spmm(const int* __restrict__ row, const int* __restrict__ col,
                       const float* __restrict__ dinv, const float* __restrict__ h,
                       float* __restrict__ agg) {
  long id = (long)blockIdx.x * blockDim.x + threadIdx.x;
  if (id < (long)NEDGES * 16) {
    int e = (int)(id >> 4);
    int q = ((int)id & 15) * 4;
    int r = row[e], c = col[e];
    float w = dinv[r] * dinv[c];
    float4 hv = *(const float4*)(h + (size_t)r * HID + q);
    float* dst = agg + (size_t)c * HID + q;
    atomAddF(dst + 0, w * hv.x);
    atomAddF(dst + 1, w * hv.y);
    atomAddF(dst + 2, w * hv.z);
    atomAddF(dst + 3, w * hv.w);
  }
}

// BN (batch stats) + ReLU, in place
__global__ void k_bn(float* __restrict__ h, const float* __restrict__ stats,
                     const float* __restrict__ gamma, const float* __restrict__ betab) {
  long i = (long)blockIdx.x * blockDim.x + threadIdx.x;
  if (i < (long)NNODES * HID) {
    int c = (int)(i & 63);
    const float invN = 1.0f / (float)NNODES;
    float mu  = stats[c] * invN;
    float var = stats[HID + c] * invN - mu * mu;
    float sc  = gamma[c] * rsqrtf(var + BN_EPS);
    float v   = (h[i] - mu) * sc + betab[c];
    h[i] = v > 0.f ? v : 0.f;
  }
}

// ---------------- WMMA GEMM kernels ----------------
// Per wave: 16 output rows x 64 cols. A 16x32 chunk staged to LDS (float4
// coalesced), B fragments read from global (weights are L0/L2 resident).
// A frag (16x4 f32): lane<16 -> A[l15][k],A[l15][k+1]; lane>=16 -> k+2,k+3.
// B frag (4x16 f32): lane<16 -> B[k][l15],B[k+1][l15]; lane>=16 -> B[k+2..3][l15].
// C/D: vgpr r, lanes0-15 -> (M=r,N=l15), lanes16-31 -> (M=r+8,N=l15).

__global__ __launch_bounds__(256) void k_gemm0(const float* __restrict__ x,
                                               const float* __restrict__ W0,
                                               const float* __restrict__ b0,
                                               float* __restrict__ h,
                                               float* __restrict__ h0) {
  __shared__ __align__(16) float As[8 * 16 * 32];
  const int lane = threadIdx.x & 31;
  const int wave = threadIdx.x >> 5;
  const int l15  = lane & 15;
  const int half = lane >> 4;
  const int rowBase = blockIdx.x * 128 + wave * 16;
  float* Aw = As + wave * 512;
  v8f acc[4] = {};
  for (int kb = 0; kb < INC; kb += 32) {
#pragma unroll
    for (int i = 0; i < 4; ++i) {            // stage 16x32 A chunk
      int flat = i * 32 + lane;
      int r = flat >> 3, c4 = (flat & 7) * 4;
      int grow = rowBase + r;
      float4 v = {0.f, 0.f, 0.f, 0.f};
      if (grow < NNODES) v = *(const float4*)(x + (size_t)grow * INC + kb + c4);
      *(float4*)(Aw + r * 32 + c4) = v;
    }
    __syncthreads();
#pragma unroll
    for (int ks = 0; ks < 32; ks += 4) {
      v2f a;
      a.x = Aw[l15 * 32 + ks + 2 * half];
      a.y = Aw[l15 * 32 + ks + 2 * half + 1];
      int kr = kb + ks + 2 * half;
#pragma unroll
      for (int t = 0; t < 4; ++t) {
        v2f b;
        b.x = W0[kr * HID + t * 16 + l15];
        b.y = W0[(kr + 1) * HID + t * 16 + l15];
        acc[t] = __builtin_amdgcn_wmma_f32_16x16x4_f32(false, a, false, b,
                                                       (short)0, acc[t], false, false);
      }
    }
    __syncthreads();
  }
#pragma unroll
  for (int t = 0; t < 4; ++t)
#pragma unroll
    for (int r = 0; r < 8; ++r) {
      int grow = rowBase + r + 8 * half;
      if (grow < NNODES) {
        int n = t * 16 + l15;
        float v = acc[t][r] + b0[n];
        v = v > 0.f ? v : 0.f;
        h[(size_t)grow * HID + n]  = v;
        h0[(size_t)grow * HID + n] = v;
      }
    }
}

// s = ((1-a)*(agg + dinv^2*h) + a*h0) @ Wp ; s written in place over h; BN stats
__global__ __launch_bounds__(256) void k_gemm_layer(const float* __restrict__ agg,
                                                    const float* hin,
                                                    const float* __restrict__ h0,
                                                    const float* __restrict__ dinv,
                                                    const float* __restrict__ Wp,
                                                    float* sout,
                                                    float* __restrict__ stats) {
  __shared__ __align__(16) float As[8 * 16 * 32];
  __shared__ float wsum[8][HID];
  __shared__ float wsq[8][HID];
  const int lane = threadIdx.x & 31;
  const int wave = threadIdx.x >> 5;
  const int l15  = lane & 15;
  const int half = lane >> 4;
  const int rowBase = blockIdx.x * 128 + wave * 16;
  float* Aw = As + wave * 512;
  v8f acc[4] = {};
  for (int kb = 0; kb < HID; kb += 32) {
#pragma unroll
    for (int i = 0; i < 4; ++i) {            // stage combined A chunk
      int flat = i * 32 + lane;
      int r = flat >> 3, c4 = (flat & 7) * 4;
      int grow = rowBase + r;
      float4 v = {0.f, 0.f, 0.f, 0.f};
      if (grow < NNODES) {
        size_t base = (size_t)grow * HID + kb + c4;
        float4 va = *(const float4*)(agg + base);
        float4 vh = *(const float4*)(hin + base);
        float4 v0 = *(const float4*)(h0 + base);
        float d = dinv[grow]; float d2 = d * d;
        v.x = OMA_F * (va.x + d2 * vh.x) + ALPHA_F * v0.x;
        v.y = OMA_F * (va.y + d2 * vh.y) + ALPHA_F * v0.y;
        v.z = OMA_F * (va.z + d2 * vh.z) + ALPHA_F * v0.z;
        v.w = OMA_F * (va.w + d2 * vh.w) + ALPHA_F * v0.w;
      }
      *(float4*)(Aw + r * 32 + c4) = v;
    }
    __syncthreads();
#pragma unroll
    for (int ks = 0; ks < 32; ks += 4) {
      v2f a;
      a.x = Aw[l15 * 32 + ks + 2 * half];
      a.y = Aw[l15 * 32 + ks + 2 * half + 1];
      int kr = kb + ks + 2 * half;
#pragma unroll
      for (int t = 0; t < 4; ++t) {
        v2f b;
        b.x = Wp[kr * HID + t * 16 + l15];
        b.y = Wp[(kr + 1) * HID + t * 16 + l15];
        acc[t] = __builtin_amdgcn_wmma_f32_16x16x4_f32(false, a, false, b,
                                                       (short)0, acc[t], false, false);
      }
    }
    __syncthreads();
  }
  // store s + wave-level BN partial sums (OOB rows have zero A -> contribute 0)
#pragma unroll
  for (int t = 0; t < 4; ++t) {
    float s8 = 0.f, q8 = 0.f;
#pragma unroll
    for (int r = 0; r < 8; ++r) {
      float v = acc[t][r];
      s8 += v; q8 += v * v;
      int grow = rowBase + r + 8 * half;
      if (grow < NNODES) sout[(size_t)grow * HID + t * 16 + l15] = v;
    }
    s8 += __shfl_xor(s8, 16, 32);
    q8 += __shfl_xor(q8, 16, 32);
    if (half == 0) { wsum[wave][t * 16 + l15] = s8; wsq[wave][t * 16 + l15] = q8; }
  }
  __syncthreads();
  if (threadIdx.x < HID) {
    float S = 0.f, Q = 0.f;
#pragma unroll
    for (int w = 0; w < 8; ++w) { S += wsum[w][threadIdx.x]; Q += wsq[w][threadIdx.x]; }
    atomAddF(&stats[threadIdx.x], S);
    atomAddF(&stats[HID + threadIdx.x], Q);
  }
}

// out = h @ W_out + b_out  (N cols = 40, 3 tiles with tail masking)
__global__ __launch_bounds__(256) void k_gemm_out(const float* __restrict__ h,
                                                  const float* __restrict__ Wout,
                                                  const float* __restrict__ bout,
                                                  float* __restrict__ out) {
  __shared__ __align__(16) float As[8 * 16 * 32];
  const int lane = threadIdx.x & 31;
  const int wave = threadIdx.x >> 5;
  const int l15  = lane & 15;
  const int half = lane >> 4;
  const int rowBase = blockIdx.x * 128 + wave * 16;
  float* Aw = As + wave * 512;
  v8f acc[3] = {};
  for (int kb = 0; kb < HID; kb += 32) {
#pragma unroll
    for (int i = 0; i < 4; ++i) {
      int flat = i * 32 + lane;
      int r = flat >> 3, c4 = (flat & 7) * 4;
      int grow = rowBase + r;
      float4 v = {0.f, 0.f, 0.f, 0.f};
      if (grow < NNODES) v = *(const float4*)(h + (size_t)grow * HID + kb + c4);
      *(float4*)(Aw + r * 32 + c4) = v;
    }
    __syncthreads();
#pragma unroll
    for (int ks = 0; ks < 32; ks += 4) {
      v2f a;
      a.x = Aw[l15 * 32 + ks + 2 * half];
      a.y = Aw[l15 * 32 + ks + 2 * half + 1];
      int kr = kb + ks + 2 * half;
#pragma unroll
      for (int t = 0; t < 3; ++t) {
        int n = t * 16 + l15;
        v2f b;
        b.x = (n < OUTC) ? Wout[kr * OUTC + n] : 0.f;
        b.y = (n < OUTC) ? Wout[(kr + 1) * OUTC + n] : 0.f;
        acc[t] = __builtin_amdgcn_wmma_f32_16x16x4_f32(false, a, false, b,
                                                       (short)0, acc[t], false, false);
      }
    }
    __syncthreads();
  }
#pragma unroll
  for (int t = 0; t < 3; ++t)
#pragma unroll
    for (int r = 0; r < 8; ++r) {
      int grow = rowBase + r + 8 * half;
      int n = t * 16 + l15;
      if (grow < NNODES && n < OUTC)
        out[(size_t)grow * OUTC + n] = acc[t][r] + bout[n];
    }
}

// ---------------- host orchestration ----------------
extern "C" void kernel_launch(void* const* d_in, const int* in_sizes, int n_in,
                              void* d_out, int out_size, void* d_ws, size_t ws_size,
                              hipStream_t stream) {
  (void)in_sizes; (void)n_in; (void)out_size; (void)ws_size;
  const float* x     = (const float*)d_in[0];
  const int*   ei    = (const int*)d_in[1];          // integer inputs -> int32
  const float* W0    = (const float*)d_in[2];
  const float* b0    = (const float*)d_in[3];
  const float* convW = (const float*)d_in[4];
  const float* gamma = (const float*)d_in[5];
  const float* betab = (const float*)d_in[6];
  const float* Wout  = (const float*)d_in[7];
  const float* bout  = (const float*)d_in[8];
  float* out = (float*)d_out;

  const long NH = (long)NNODES * HID;
  float* ws     = (float*)d_ws;
  float* hbuf   = ws;
  float* h0buf  = ws + NH;
  float* aggbuf = ws + 2 * NH;
  float* dinv   = ws + 3 * NH;
  float* Wp     = dinv + NNODES;
  float* stats  = Wp + HID * HID;

  const int* erow = ei;
  const int* ecol = ei + NEDGES;

  const int TB = 256;
  const int GB = (NNODES + 127) / 128;               // gemm blocks (8 waves x 16 rows)

  // degrees (self loop contributes 1) -> dinv = deg^-1/2
  k_fill<<<(NNODES + TB - 1) / TB, TB, 0, stream>>>(dinv, NNODES, 1.0f);
  k_degcount<<<(NEDGES + TB - 1) / TB, TB, 0, stream>>>(ecol, dinv);
  k_rsqrt<<<(NNODES + TB - 1) / TB, TB, 0, stream>>>(dinv);

  // h = h0 = relu(x @ W0 + b0)
  k_gemm0<<<GB, TB, 0, stream>>>(x, W0, b0, hbuf, h0buf);

  for (int l = 0; l < NLAYERS; ++l) {
    float beta = (float)log(0.5 / (double)(l + 1) + 1.0);
    k_fill<<<(int)((NH + TB - 1) / TB), TB, 0, stream>>>(aggbuf, NH, 0.0f);
    k_spmm<<<(int)(((long)NEDGES * 16 + TB - 1) / TB), TB, 0, stream>>>(
        erow, ecol, dinv, hbuf, aggbuf);
    k_wp<<<(HID * HID + TB - 1) / TB, TB, 0, stream>>>(convW + (size_t)l * HID * HID,
                                                       Wp, beta, stats);
    k_gemm_layer<<<GB, TB, 0, stream>>>(aggbuf, hbuf, h0buf, dinv, Wp, hbuf, stats);
    k_bn<<<(int)((NH + TB - 1) / TB), TB, 0, stream>>>(hbuf, stats,
                                                       gamma + l * HID, betab + l * HID);
  }

  k_gemm_out<<<GB, TB, 0, stream>>>(hbuf, Wout, bout, out);
}